// GCN_38293928411392
// MI455X (gfx1250) — compile-verified
//
#include <hip/hip_runtime.h>
#include <hip/hip_bf16.h>
#include <math.h>
#include <stdint.h>

#define N_NODES 100000
#define N_EDGES 1600000
#define N_GRAPHS 1024
#define DF 32
#define NC 10
#define TILES (N_NODES / 16)
#define WT_STRIDE 34  // transposed-weight row stride (even: keeps b64 alignment, spreads banks)

typedef __attribute__((ext_vector_type(2))) float v2f;
typedef __attribute__((ext_vector_type(8))) float v8f;

// ---------------- CDNA5 async global->LDS staging ----------------
// GLOBAL_LOAD_ASYNC_TO_LDS_B128 (ISA 15.18.3 op 98): per-lane 16-byte copy from global memory to LDS,
// tracked by ASYNCcnt. LDS byte address = low 32 bits of the generic shared pointer.
static __device__ __forceinline__ void async_stage_b128(float* lds_dst, const float* g_src)
{
    uint32_t loff = (uint32_t)(uintptr_t)lds_dst;
    asm volatile("global_load_async_to_lds_b128 %0, %1, off"
                 :: "v"(loff), "v"(g_src) : "memory");
}

static __device__ __forceinline__ void wait_async0()
{
#if __has_builtin(__builtin_amdgcn_s_wait_asynccnt)
    __builtin_amdgcn_s_wait_asynccnt(0);
#else
    asm volatile("s_wait_asynccnt 0" ::: "memory");
#endif
}

// stage n4 float4's from global to LDS cooperatively (call from all threads)
static __device__ __forceinline__ void stage_weights(float* lds_dst, const float* g_src,
                                                     int n4, int tid, int nthreads)
{
    for (int i = tid; i < n4; i += nthreads)
        async_stage_b128(lds_dst + i * 4, g_src + i * 4);
}

// transpose raw row-major W (32 x cols) in LDS into Wt (cols rows x 32 k, stride WT_STRIDE)
static __device__ __forceinline__ void transpose_w(const float* __restrict__ raw,
                                                   float* __restrict__ wt, int cols,
                                                   int tid, int nthreads)
{
    for (int idx = tid; idx < 32 * cols; idx += nthreads) {
        int k = idx / cols, n = idx - k * cols;
        wt[n * WT_STRIDE + k] = raw[idx];
    }
}

// ---------------- WMMA helpers (V_WMMA_F32_16X16X4_F32, f32-exact) ----------------
// Layouts per ISA 7.12.2 (32-bit): A lane l holds A[M=l&15][K=2*(l>>4)+j] in VGPR j;
// B lane l holds B[K=2*(l>>4)+j][N=n0+(l&15)]; C/D lane l VGPR r = D[M=r+8*(l>>4)][N=n0+(l&15)].
struct AFrag { v2f a[8]; };

static __device__ __forceinline__ AFrag ld_afrag(const float* __restrict__ A, int lane)
{
    AFrag f;
    const int m = lane & 15, g = lane >> 4;
#pragma unroll
    for (int c = 0; c < 8; ++c)
        f.a[c] = *(const v2f*)(A + m * DF + c * 4 + g * 2);  // contiguous K pair, 8B aligned
    return f;
}

// acc = A(16x32) * B(32 x N, col tile n0) + acc ; B from transposed LDS: one b64 per frag
static __device__ __forceinline__ v8f wmma_chain(const AFrag& f, const float* __restrict__ Wt,
                                                 int n0, v8f acc, int lane)
{
    const float* col = Wt + (n0 + (lane & 15)) * WT_STRIDE + (lane >> 4) * 2;
    v2f b[8];
#pragma unroll
    for (int c = 0; c < 8; ++c)
        b[c] = *(const v2f*)(col + c * 4);  // contiguous (ka, ka+1), 8B aligned
#pragma unroll
    for (int c = 0; c < 8; ++c)
        acc = __builtin_amdgcn_wmma_f32_16x16x4_f32(false, f.a[c], false, b[c], (short)0,
                                                    acc, false, false);
    return acc;
}

static __device__ __forceinline__ void st_tile(float* __restrict__ O, int cols, int n0,
                                               v8f d, int lane)
{
    const int m = lane & 15, g = lane >> 4, n = n0 + m;
#pragma unroll
    for (int r = 0; r < 8; ++r) O[(r + 8 * g) * cols + n] = d[r];
}

static __device__ __forceinline__ v8f ld_tile(const float* __restrict__ I, int cols, int n0,
                                              int lane)
{
    v8f c;
    const int m = lane & 15, g = lane >> 4, n = n0 + m;
#pragma unroll
    for (int r = 0; r < 8; ++r) c[r] = I[(r + 8 * g) * cols + n];
    return c;
}

static __device__ __forceinline__ float sigmoidf_(float x) { return 1.0f / (1.0f + __expf(-x)); }

// ---------------- conv1 (F_IN = 1): k/q/v are outer products ----------------
__global__ void k_conv1_node(const float* __restrict__ x,
                             const float* __restrict__ Wk, const float* __restrict__ bk,
                             const float* __restrict__ Wq, const float* __restrict__ bq,
                             const float* __restrict__ Wv, const float* __restrict__ bv,
                             float* __restrict__ K, float* __restrict__ Q, float* __restrict__ V)
{
    int tid = blockIdx.x * blockDim.x + threadIdx.x;
    if (tid >= N_NODES * DF) return;
    int i = tid >> 5, d = tid & 31;
    float xi = x[i];
    K[tid] = xi * Wk[d] + bk[d];
    Q[tid] = xi * Wq[d] + bq[d];
    V[tid] = xi * Wv[d] + bv[d];
}

// ---------------- ResGated edge pass: sigmoid(k[dst]+q[src])*v[src] -> agg[dst] --------
__global__ void k_rg_edge(const int* __restrict__ ei,
                          const float* __restrict__ K, const float* __restrict__ Q,
                          const float* __restrict__ V, float* __restrict__ agg)
{
    long tid = (long)blockIdx.x * blockDim.x + threadIdx.x;
    if (tid >= (long)N_EDGES * 8) return;
    int e = (int)(tid >> 3), sub = (int)(tid & 7);
    int s = ei[e], d = ei[N_EDGES + e];
    float4 kd = ((const float4*)K)[d * 8 + sub];
    float4 qs = ((const float4*)Q)[s * 8 + sub];
    float4 vs = ((const float4*)V)[s * 8 + sub];
    float4 m;
    m.x = vs.x * sigmoidf_(kd.x + qs.x);
    m.y = vs.y * sigmoidf_(kd.y + qs.y);
    m.z = vs.z * sigmoidf_(kd.z + qs.z);
    m.w = vs.w * sigmoidf_(kd.w + qs.w);
    float* a = agg + (size_t)d * DF + sub * 4;
    atomicAdd(a + 0, m.x);
    atomicAdd(a + 1, m.y);
    atomicAdd(a + 2, m.z);
    atomicAdd(a + 3, m.w);
}

__global__ void k_conv1_post(const float* __restrict__ x, const float* __restrict__ agg,
                             const float* __restrict__ Wsk, const float* __restrict__ b,
                             float* __restrict__ h1)
{
    int tid = blockIdx.x * blockDim.x + threadIdx.x;
    if (tid >= N_NODES * DF) return;
    int i = tid >> 5, d = tid & 31;
    float v = agg[tid] + x[i] * Wsk[d] + b[d];
    h1[tid] = fmaxf(v, 0.0f);
}

// ---------------- FiLM node-side: out_skip, xl, beta, gamma (all via WMMA) ------------
// One wave handles one (row-tile, column-tile) pair: low register pressure, no spills.
__global__ void k_film_node(const float* __restrict__ h1,
                            const float* __restrict__ Wlin, const float* __restrict__ Wfilm,
                            const float* __restrict__ bfilm,
                            const float* __restrict__ Wls, const float* __restrict__ Wfs,
                            float* __restrict__ outskip, float* __restrict__ xl,
                            float* __restrict__ beta, float* __restrict__ gamma)
{
    // raw row-major images (6144) then transposed (2*32*34 + 2*64*34 = 6528)
    __shared__ float sW[6144 + 6528];
    float* raw      = sW;            // Wlin | Wls | Wfilm | Wfs
    float* wt_lin   = sW + 6144;                 // 32*34
    float* wt_ls    = wt_lin + 32 * WT_STRIDE;   // 32*34
    float* wt_film  = wt_ls + 32 * WT_STRIDE;    // 64*34
    float* wt_fs    = wt_film + 64 * WT_STRIDE;  // 64*34
    const int tid = threadIdx.x;
    stage_weights(raw + 0,    Wlin,  256, tid, blockDim.x);
    stage_weights(raw + 1024, Wls,   256, tid, blockDim.x);
    stage_weights(raw + 2048, Wfilm, 512, tid, blockDim.x);
    stage_weights(raw + 4096, Wfs,   512, tid, blockDim.x);
    wait_async0();
    __syncthreads();
    transpose_w(raw + 0,    wt_lin,  32, tid, blockDim.x);
    transpose_w(raw + 1024, wt_ls,   32, tid, blockDim.x);
    transpose_w(raw + 2048, wt_film, 64, tid, blockDim.x);
    transpose_w(raw + 4096, wt_fs,   64, tid, blockDim.x);
    __syncthreads();

    const int gwave = (blockIdx.x * blockDim.x + threadIdx.x) >> 5;
    const int lane = threadIdx.x & 31;
    const int tile = gwave >> 1;
    const int t = gwave & 1;          // column tile (0 or 1)
    if (tile >= TILES) return;        // wave-uniform: EXEC stays all-ones for WMMA
    const float* A = h1 + (size_t)tile * 16 * DF;
    const size_t ro = (size_t)tile * 16 * DF;
    const AFrag f = ld_afrag(A, lane);
    const int n = t * 16 + (lane & 15);
    const v8f z = {0.f, 0.f, 0.f, 0.f, 0.f, 0.f, 0.f, 0.f};

    // skip path: relu(gamma_s * (h@Wls) + beta_s), split of h@Wfs
    v8f ls = wmma_chain(f, wt_ls, t * 16, z, lane);
    v8f bs = wmma_chain(f, wt_fs, t * 16, z, lane);
    v8f gs = wmma_chain(f, wt_fs, DF + t * 16, z, lane);
    v8f os;
#pragma unroll
    for (int r = 0; r < 8; ++r) os[r] = fmaxf(gs[r] * ls[r] + bs[r], 0.0f);
    st_tile(outskip + ro, DF, t * 16, os, lane);

    // message path precompute: xl = h@Wlin ; [beta|gamma] = h@Wfilm + bfilm
    v8f xv = wmma_chain(f, wt_lin, t * 16, z, lane);
    st_tile(xl + ro, DF, t * 16, xv, lane);
    v8f be = wmma_chain(f, wt_film, t * 16, z, lane);
    v8f ga = wmma_chain(f, wt_film, DF + t * 16, z, lane);
    const float bb = bfilm[n], gb = bfilm[DF + n];
#pragma unroll
    for (int r = 0; r < 8; ++r) { be[r] += bb; ga[r] += gb; }
    st_tile(beta + ro, DF, t * 16, be, lane);
    st_tile(gamma + ro, DF, t * 16, ga, lane);
}

// ---------------- FiLM edge pass: relu(gamma[dst]*xl[src]+beta[dst]) -> sums, cnt -----
__global__ void k_film_edge(const int* __restrict__ ei, const float* __restrict__ xl,
                            const float* __restrict__ beta, const float* __restrict__ gamma,
                            float* __restrict__ sums, float* __restrict__ cnt)
{
    long tid = (long)blockIdx.x * blockDim.x + threadIdx.x;
    if (tid >= (long)N_EDGES * 8) return;
    int e = (int)(tid >> 3), sub = (int)(tid & 7);
    int s = ei[e], d = ei[N_EDGES + e];
    float4 xv = ((const float4*)xl)[s * 8 + sub];
    float4 be = ((const float4*)beta)[d * 8 + sub];
    float4 ga = ((const float4*)gamma)[d * 8 + sub];
    float4 m;
    m.x = fmaxf(ga.x * xv.x + be.x, 0.0f);
    m.y = fmaxf(ga.y * xv.y + be.y, 0.0f);
    m.z = fmaxf(ga.z * xv.z + be.z, 0.0f);
    m.w = fmaxf(ga.w * xv.w + be.w, 0.0f);
    float* a = sums + (size_t)d * DF + sub * 4;
    atomicAdd(a + 0, m.x);
    atomicAdd(a + 1, m.y);
    atomicAdd(a + 2, m.z);
    atomicAdd(a + 3, m.w);
    if (sub == 0) atomicAdd(cnt + d, 1.0f);
}

__global__ void k_film_post(const float* __restrict__ outskip, const float* __restrict__ sums,
                            const float* __restrict__ cnt, float* __restrict__ h2)
{
    int tid = blockIdx.x * blockDim.x + threadIdx.x;
    if (tid >= N_NODES * DF) return;
    int i = tid >> 5;
    float c = fmaxf(cnt[i], 1.0f);
    h2[tid] = fmaxf(outskip[tid] + sums[tid] / c, 0.0f);
}

// ---------------- conv3 node-side: k/q/v = h2@W + b (WMMA) ----------------
__global__ void k_conv3_kqv(const float* __restrict__ h2,
                            const float* __restrict__ Wk, const float* __restrict__ bk,
                            const float* __restrict__ Wq, const float* __restrict__ bq,
                            const float* __restrict__ Wv, const float* __restrict__ bv,
                            float* __restrict__ K, float* __restrict__ Q, float* __restrict__ V)
{
    __shared__ float sW[3072 + 3 * 32 * WT_STRIDE];
    float* raw   = sW;
    float* wt_k  = sW + 3072;
    float* wt_q  = wt_k + 32 * WT_STRIDE;
    float* wt_v  = wt_q + 32 * WT_STRIDE;
    const int tid = threadIdx.x;
    stage_weights(raw + 0,    Wk, 256, tid, blockDim.x);
    stage_weights(raw + 1024, Wq, 256, tid, blockDim.x);
    stage_weights(raw + 2048, Wv, 256, tid, blockDim.x);
    wait_async0();
    __syncthreads();
    transpose_w(raw + 0,    wt_k, 32, tid, blockDim.x);
    transpose_w(raw + 1024, wt_q, 32, tid, blockDim.x);
    transpose_w(raw + 2048, wt_v, 32, tid, blockDim.x);
    __syncthreads();

    const int gwave = (blockIdx.x * blockDim.x + threadIdx.x) >> 5;
    const int lane = threadIdx.x & 31;
    const int tile = gwave >> 1;
    const int t = gwave & 1;
    if (tile >= TILES) return;
    const float* A = h2 + (size_t)tile * 16 * DF;
    const size_t ro = (size_t)tile * 16 * DF;
    const AFrag f = ld_afrag(A, lane);
    const int n = t * 16 + (lane & 15);
    const v8f z = {0.f, 0.f, 0.f, 0.f, 0.f, 0.f, 0.f, 0.f};

    v8f kk = wmma_chain(f, wt_k, t * 16, z, lane);
    v8f qq = wmma_chain(f, wt_q, t * 16, z, lane);
    v8f vv = wmma_chain(f, wt_v, t * 16, z, lane);
    const float kb = bk[n], qb = bq[n], vb = bv[n];
#pragma unroll
    for (int r = 0; r < 8; ++r) { kk[r] += kb; qq[r] += qb; vv[r] += vb; }
    st_tile(K + ro, DF, t * 16, kk, lane);
    st_tile(Q + ro, DF, t * 16, qq, lane);
    st_tile(V + ro, DF, t * 16, vv, lane);
}

// ---------------- conv3 post: h3 = agg + h2@Wskip + b (WMMA, C init from agg) ---------
__global__ void k_conv3_post(const float* __restrict__ h2, const float* __restrict__ agg,
                             const float* __restrict__ Wsk, const float* __restrict__ b,
                             float* __restrict__ h3)
{
    __shared__ float sW[1024 + 32 * WT_STRIDE];
    float* raw  = sW;
    float* wt_s = sW + 1024;
    stage_weights(raw, Wsk, 256, threadIdx.x, blockDim.x);
    wait_async0();
    __syncthreads();
    transpose_w(raw, wt_s, 32, threadIdx.x, blockDim.x);
    __syncthreads();

    const int gwave = (blockIdx.x * blockDim.x + threadIdx.x) >> 5;
    const int lane = threadIdx.x & 31;
    const int tile = gwave >> 1;
    const int t = gwave & 1;
    if (tile >= TILES) return;
    const float* A = h2 + (size_t)tile * 16 * DF;
    const size_t ro = (size_t)tile * 16 * DF;
    const AFrag f = ld_afrag(A, lane);
    const int n = t * 16 + (lane & 15);

    v8f acc = ld_tile(agg + ro, DF, t * 16, lane);
    acc = wmma_chain(f, wt_s, t * 16, acc, lane);
    const float bb = b[n];
#pragma unroll
    for (int r = 0; r < 8; ++r) acc[r] += bb;
    st_tile(h3 + ro, DF, t * 16, acc, lane);
}

// ---------------- global mean pool + classifier ----------------
__global__ void k_pool(const float* __restrict__ h3, const int* __restrict__ batch,
                       float* __restrict__ psum, float* __restrict__ pcnt)
{
    int tid = blockIdx.x * blockDim.x + threadIdx.x;
    if (tid >= N_NODES * 8) return;
    int i = tid >> 3, sub = tid & 7;
    int g = batch[i];
    float4 hv = ((const float4*)h3)[i * 8 + sub];
    float* p = psum + (size_t)g * DF + sub * 4;
    atomicAdd(p + 0, hv.x);
    atomicAdd(p + 1, hv.y);
    atomicAdd(p + 2, hv.z);
    atomicAdd(p + 3, hv.w);
    if (sub == 0) atomicAdd(pcnt + g, 1.0f);
}

__global__ void k_classify(const float* __restrict__ psum, const float* __restrict__ pcnt,
                           const float* __restrict__ W, const float* __restrict__ b,
                           float* __restrict__ out)
{
    int g = blockIdx.x * blockDim.x + threadIdx.x;
    if (g >= N_GRAPHS) return;
    float inv = 1.0f / fmaxf(pcnt[g], 1.0f);
    float acc[NC];
#pragma unroll
    for (int c = 0; c < NC; ++c) acc[c] = b[c];
    for (int d = 0; d < DF; ++d) {
        float p = psum[(size_t)g * DF + d] * inv;
#pragma unroll
        for (int c = 0; c < NC; ++c) acc[c] += p * W[d * NC + c];
    }
#pragma unroll
    for (int c = 0; c < NC; ++c) out[g * NC + c] = acc[c];
}

extern "C" void kernel_launch(void* const* d_in, const int* in_sizes, int n_in,
                              void* d_out, int out_size, void* d_ws, size_t ws_size,
                              hipStream_t stream)
{
    const float* x        = (const float*)d_in[0];
    const int*   ei       = (const int*)d_in[1];
    const int*   batch    = (const int*)d_in[2];
    const float* c1_Wk    = (const float*)d_in[4];
    const float* c1_bk    = (const float*)d_in[5];
    const float* c1_Wq    = (const float*)d_in[6];
    const float* c1_bq    = (const float*)d_in[7];
    const float* c1_Wv    = (const float*)d_in[8];
    const float* c1_bv    = (const float*)d_in[9];
    const float* c1_Wsk   = (const float*)d_in[10];
    const float* c1_b     = (const float*)d_in[11];
    const float* c2_Wlin  = (const float*)d_in[12];
    const float* c2_Wfilm = (const float*)d_in[13];
    const float* c2_bfilm = (const float*)d_in[14];
    const float* c2_Wls   = (const float*)d_in[15];
    const float* c2_Wfs   = (const float*)d_in[16];
    const float* c3_Wk    = (const float*)d_in[17];
    const float* c3_bk    = (const float*)d_in[18];
    const float* c3_Wq    = (const float*)d_in[19];
    const float* c3_bq    = (const float*)d_in[20];
    const float* c3_Wv    = (const float*)d_in[21];
    const float* c3_bv    = (const float*)d_in[22];
    const float* c3_Wsk   = (const float*)d_in[23];
    const float* c3_b     = (const float*)d_in[24];
    const float* linW     = (const float*)d_in[25];
    const float* linb     = (const float*)d_in[26];

    float* ws = (float*)d_ws;
    const size_t S = (size_t)N_NODES * DF;
    float* s0 = ws + 0 * S;   // k / out_skip / k3
    float* s1 = ws + 1 * S;   // q / xl / q3
    float* s2 = ws + 2 * S;   // v / beta / v3
    float* s3 = ws + 3 * S;   // agg / sums
    float* s4 = ws + 4 * S;   // h1, later h3
    float* s5 = ws + 5 * S;   // gamma
    float* s6 = ws + 6 * S;   // h2
    float* cnt  = ws + 7 * S;                    // N floats
    float* psum = cnt + N_NODES;                 // G*32
    float* pcnt = psum + (size_t)N_GRAPHS * DF;  // G

    const int TB = 256;
    const int nd_blocks = (N_NODES * DF + TB - 1) / TB;
    const long eth = (long)N_EDGES * 8;
    const int e_blocks = (int)((eth + TB - 1) / TB);
    const int w2_blocks = (TILES * 2 * 32 + TB - 1) / TB;  // (tile, col-tile) waves
    const int p_blocks = (N_NODES * 8 + TB - 1) / TB;

    // ---- conv1: ResGated(F_IN=1 -> 32) ----
    k_conv1_node<<<nd_blocks, TB, 0, stream>>>(x, c1_Wk, c1_bk, c1_Wq, c1_bq, c1_Wv, c1_bv,
                                               s0, s1, s2);
    hipMemsetAsync(s3, 0, S * sizeof(float), stream);
    k_rg_edge<<<e_blocks, TB, 0, stream>>>(ei, s0, s1, s2, s3);
    k_conv1_post<<<nd_blocks, TB, 0, stream>>>(x, s3, c1_Wsk, c1_b, s4);  // h1 = relu(...)

    // ---- conv2: FiLM(32 -> 32), mean aggregation ----
    k_film_node<<<w2_blocks, TB, 0, stream>>>(s4, c2_Wlin, c2_Wfilm, c2_bfilm, c2_Wls, c2_Wfs,
                                              s0, s1, s2, s5);
    hipMemsetAsync(s3, 0, S * sizeof(float), stream);
    hipMemsetAsync(cnt, 0, N_NODES * sizeof(float), stream);
    k_film_edge<<<e_blocks, TB, 0, stream>>>(ei, s1, s2, s5, s3, cnt);
    k_film_post<<<nd_blocks, TB, 0, stream>>>(s0, s3, cnt, s6);  // h2 = relu(...)

    // ---- conv3: ResGated(32 -> 32) ----
    k_conv3_kqv<<<w2_blocks, TB, 0, stream>>>(s6, c3_Wk, c3_bk, c3_Wq, c3_bq, c3_Wv, c3_bv,
                                              s0, s1, s2);
    hipMemsetAsync(s3, 0, S * sizeof(float), stream);
    k_rg_edge<<<e_blocks, TB, 0, stream>>>(ei, s0, s1, s2, s3);
    k_conv3_post<<<w2_blocks, TB, 0, stream>>>(s6, s3, c3_Wsk, c3_b, s4);  // h3 (no relu)

    // ---- global mean pool + classifier ----
    hipMemsetAsync(psum, 0, (size_t)N_GRAPHS * DF * sizeof(float), stream);
    hipMemsetAsync(pcnt, 0, N_GRAPHS * sizeof(float), stream);
    k_pool<<<p_blocks, TB, 0, stream>>>(s4, batch, psum, pcnt);
    k_classify<<<(N_GRAPHS + 63) / 64, 64, 0, stream>>>(psum, pcnt, linW, linb, (float*)d_out);
}